// GNN_89661737271610
// MI455X (gfx1250) — compile-verified
//
#include <hip/hip_runtime.h>

typedef float v2f __attribute__((ext_vector_type(2)));
typedef float v8f __attribute__((ext_vector_type(8)));

static constexpr int kNodes  = 100000;
static constexpr int kEdges  = 1600000;
static constexpr int kInDim  = 256;
static constexpr int kHid    = 64;
static constexpr int kGraphs = 100;

// ---------------------------------------------------------------------------
// utilities
// ---------------------------------------------------------------------------
__global__ void gcn_fill_f32(float* __restrict__ p, float v, int n) {
    int i = blockIdx.x * blockDim.x + threadIdx.x;
    if (i < n) p[i] = v;
}

__global__ void gcn_fill_i32(int* __restrict__ p, int v, int n) {
    int i = blockIdx.x * blockDim.x + threadIdx.x;
    if (i < n) p[i] = v;
}

// degi[d] += 1 for every real edge (degi pre-initialized to 1 for self loop)
__global__ void gcn_deg_count(const int* __restrict__ dst, int* __restrict__ degi, int E) {
    int e = blockIdx.x * blockDim.x + threadIdx.x;
    if (e < E) atomicAdd(&degi[dst[e]], 1);
}

__global__ void gcn_dinv(const int* __restrict__ degi, float* __restrict__ dinv, int n) {
    int i = blockIdx.x * blockDim.x + threadIdx.x;
    if (i < n) {
        float d = (float)degi[i];
        dinv[i] = (d > 0.0f) ? rsqrtf(d) : 0.0f;
    }
}

// ---------------------------------------------------------------------------
// Single-workgroup exclusive scan over degi -> rowptr (and cursor copy).
// ---------------------------------------------------------------------------
__global__ __launch_bounds__(1024) void gcn_scan(const int* __restrict__ degi,
                                                 int* __restrict__ rowptr,
                                                 int* __restrict__ cursor, int n) {
    __shared__ int partial[1024];
    const int t = threadIdx.x;
    const int chunk = (n + 1023) / 1024;
    const int lo = t * chunk;
    const int hi = (lo + chunk < n) ? lo + chunk : n;

    int s = 0;
    for (int i = lo; i < hi; ++i) s += degi[i];
    partial[t] = s;
    __syncthreads();

    for (int off = 1; off < 1024; off <<= 1) {
        int v = (t >= off) ? partial[t - off] : 0;
        __syncthreads();
        partial[t] += v;
        __syncthreads();
    }

    int base = partial[t] - s;   // exclusive prefix of this thread's chunk
    for (int i = lo; i < hi; ++i) {
        rowptr[i] = base;
        cursor[i] = base;
        base += degi[i];
    }
    if (t == 1023) rowptr[n] = partial[1023];
}

// ---------------------------------------------------------------------------
// CSR fill: for every edge (incl. trailing self loops) grab a slot in its
// destination row; bake src index + normalization weight into the slot.
// ---------------------------------------------------------------------------
__global__ void gcn_csr_fill(const int* __restrict__ src, const int* __restrict__ dst,
                             const float* __restrict__ dinv, int* __restrict__ cursor,
                             int* __restrict__ esrc, float* __restrict__ enorm,
                             int E, int total) {
    int e = blockIdx.x * blockDim.x + threadIdx.x;
    if (e >= total) return;
    int s, d; float w;
    if (e < E) {
        s = src[e]; d = dst[e];
        w = dinv[s] * dinv[d];
    } else {
        s = e - E; d = s;
        float di = dinv[s];
        w = di * di;
    }
    int slot = atomicAdd(&cursor[d], 1);
    esrc[slot] = s;
    enorm[slot] = w;
}

// ---------------------------------------------------------------------------
// GEMM: C[M x 64] = A[M x K] * B[K x 64] via V_WMMA_F32_16X16X4_F32.
// One wave computes a 16x64 tile (4 accumulators). M % 16 == 0.
// ---------------------------------------------------------------------------
__global__ __launch_bounds__(256) void gcn_gemm_n64_wmma(
    const float* __restrict__ A, const float* __restrict__ B,
    float* __restrict__ C, int M, int K) {
    const int lane = threadIdx.x & 31;
    const int wave = (blockIdx.x * blockDim.x + threadIdx.x) >> 5;
    const int m0 = wave << 4;
    if (m0 >= M) return;                  // wave-uniform: EXEC stays all-1s
    const int half = lane >> 4;
    const int l16  = lane & 15;
    const long arow = (long)(m0 + l16) * (long)K;

    v8f acc0 = {}, acc1 = {}, acc2 = {}, acc3 = {};
    for (int kk = 0; kk < K; kk += 4) {
        const int ka = kk + 2 * half;
        v2f a;
        a.x = A[arow + ka];
        a.y = A[arow + ka + 1];
        const float* __restrict__ Br0 = B + (long)ka * kHid;
        const float* __restrict__ Br1 = Br0 + kHid;
        v2f b0; b0.x = Br0[l16];      b0.y = Br1[l16];
        v2f b1; b1.x = Br0[16 + l16]; b1.y = Br1[16 + l16];
        v2f b2; b2.x = Br0[32 + l16]; b2.y = Br1[32 + l16];
        v2f b3; b3.x = Br0[48 + l16]; b3.y = Br1[48 + l16];
        acc0 = __builtin_amdgcn_wmma_f32_16x16x4_f32(false, a, false, b0, (short)0, acc0, false, false);
        acc1 = __builtin_amdgcn_wmma_f32_16x16x4_f32(false, a, false, b1, (short)0, acc1, false, false);
        acc2 = __builtin_amdgcn_wmma_f32_16x16x4_f32(false, a, false, b2, (short)0, acc2, false, false);
        acc3 = __builtin_amdgcn_wmma_f32_16x16x4_f32(false, a, false, b3, (short)0, acc3, false, false);
    }

#pragma unroll
    for (int r = 0; r < 8; ++r) {
        const long row = (long)(m0 + r + 8 * half);
        float* __restrict__ Crow = C + row * kHid;
        Crow[ 0 + l16] = acc0[r];
        Crow[16 + l16] = acc1[r];
        Crow[32 + l16] = acc2[r];
        Crow[48 + l16] = acc3[r];
    }
}

// ---------------------------------------------------------------------------
// Fused CSR aggregation + bias + activation: one wave per destination node.
// ---------------------------------------------------------------------------
__global__ void gcn_agg_csr(const int* __restrict__ rowptr, const int* __restrict__ esrc,
                            const float* __restrict__ enorm, const float* __restrict__ h,
                            const float* __restrict__ bias, float* __restrict__ out,
                            int n, int relu) {
    const int lane = threadIdx.x & 31;
    const int node = (blockIdx.x * blockDim.x + threadIdx.x) >> 5;
    if (node >= n) return;                // wave-uniform
    const int j0 = rowptr[node];
    const int j1 = rowptr[node + 1];
    float a0 = 0.0f, a1 = 0.0f;
    for (int j = j0; j < j1; ++j) {
        const int s = esrc[j];
        const float w = enorm[j];
        const float* __restrict__ hs = h + (long)s * kHid;
        a0 = fmaf(hs[lane],      w, a0);
        a1 = fmaf(hs[lane + 32], w, a1);
    }
    float v0 = a0 + bias[lane];
    float v1 = a1 + bias[lane + 32];
    if (relu) { v0 = fmaxf(v0, 0.0f); v1 = fmaxf(v1, 0.0f); }
    float* __restrict__ o = out + (long)node * kHid;
    o[lane]      = v0;
    o[lane + 32] = v1;
}

// ---------------------------------------------------------------------------
// Segmented pooling: batch is SORTED, so each wave reduces a contiguous chunk
// of nodes in registers and atomically flushes only on (wave-uniform) graph
// boundary crossings. ~16-32x fewer atomics than per-element scatter.
// ---------------------------------------------------------------------------
static constexpr int kPoolChunk = 32;     // nodes per wave

__global__ void gcn_pool_seg(const float* __restrict__ h, const int* __restrict__ batch,
                             float* __restrict__ pooled, int n) {
    const int lane = threadIdx.x & 31;
    const int c = (blockIdx.x * blockDim.x + threadIdx.x) >> 5;
    const int i0 = c * kPoolChunk;
    if (i0 >= n) return;                  // wave-uniform
    const int i1 = (i0 + kPoolChunk < n) ? i0 + kPoolChunk : n;

    int curg = batch[i0];                 // wave-uniform (all lanes same addr)
    float a0 = 0.0f, a1 = 0.0f;
    for (int i = i0; i < i1; ++i) {
        const int g = batch[i];
        if (g != curg) {                  // wave-uniform branch
            atomicAdd(&pooled[(long)curg * kHid + lane],      a0);
            atomicAdd(&pooled[(long)curg * kHid + lane + 32], a1);
            a0 = 0.0f; a1 = 0.0f; curg = g;
        }
        const float* __restrict__ hi = h + (long)i * kHid;
        a0 += hi[lane];
        a1 += hi[lane + 32];
    }
    atomicAdd(&pooled[(long)curg * kHid + lane],      a0);
    atomicAdd(&pooled[(long)curg * kHid + lane + 32], a1);
}

// out[g] = dot(pooled[g], Wl) + bl
__global__ void gcn_head(const float* __restrict__ pooled, const float* __restrict__ Wl,
                         const float* __restrict__ bl, float* __restrict__ out, int G) {
    int g = blockIdx.x * blockDim.x + threadIdx.x;
    if (g < G) {
        float s = 0.0f;
#pragma unroll
        for (int f = 0; f < kHid; ++f) s += pooled[g * kHid + f] * Wl[f];
        out[g] = s + bl[0];
    }
}

extern "C" void kernel_launch(void* const* d_in, const int* in_sizes, int n_in,
                              void* d_out, int out_size, void* d_ws, size_t ws_size,
                              hipStream_t stream) {
    (void)in_sizes; (void)n_in; (void)out_size; (void)ws_size;
    const float* x   = (const float*)d_in[0];
    const int*   ei  = (const int*)d_in[1];
    const int*   bat = (const int*)d_in[2];
    const float* W1  = (const float*)d_in[3];
    const float* b1  = (const float*)d_in[4];
    const float* W2  = (const float*)d_in[5];
    const float* b2  = (const float*)d_in[6];
    const float* W3  = (const float*)d_in[7];
    const float* b3  = (const float*)d_in[8];
    const float* Wl  = (const float*)d_in[9];
    const float* bl  = (const float*)d_in[10];
    float* out = (float*)d_out;

    const int* src = ei;
    const int* dst = ei + kEdges;
    const int total = kEdges + kNodes;          // edges + self loops
    const int nhid  = kNodes * kHid;            // 6.4M

    // workspace carve-out (256B aligned)
    char* ws = (char*)d_ws;
    size_t off = 0;
    auto carve = [&](size_t bytes) -> void* {
        void* p = (void*)(ws + off);
        off = (off + bytes + 255) & ~(size_t)255;
        return p;
    };
    int*   degi   = (int*)  carve((size_t)kNodes * 4);
    int*   rowptr = (int*)  carve((size_t)(kNodes + 1) * 4);
    int*   cursor = (int*)  carve((size_t)kNodes * 4);
    float* dinv   = (float*)carve((size_t)kNodes * 4);
    int*   esrc   = (int*)  carve((size_t)total * 4);
    float* enorm  = (float*)carve((size_t)total * 4);
    float* bufA   = (float*)carve((size_t)nhid * 4);
    float* bufB   = (float*)carve((size_t)nhid * 4);
    float* pooled = (float*)carve((size_t)kGraphs * kHid * 4);

    auto cdiv = [](int a, int b) { return (a + b - 1) / b; };
    const int gemmBlocks = cdiv(kNodes / 16, 8);          // 1 wave = 16 rows
    const int aggBlocks  = cdiv(kNodes, 8);               // 1 wave = 1 node
    const int poolBlocks = cdiv(cdiv(kNodes, kPoolChunk), 8);

    // --- CSR + normalization (rebuilt every call; ws is not trusted) ---
    gcn_fill_i32 <<<cdiv(kNodes, 256), 256, 0, stream>>>(degi, 1, kNodes);   // self loop
    gcn_deg_count<<<cdiv(kEdges, 256), 256, 0, stream>>>(dst, degi, kEdges);
    gcn_dinv     <<<cdiv(kNodes, 256), 256, 0, stream>>>(degi, dinv, kNodes);
    gcn_scan     <<<1, 1024, 0, stream>>>(degi, rowptr, cursor, kNodes);
    gcn_csr_fill <<<cdiv(total, 256), 256, 0, stream>>>(src, dst, dinv, cursor,
                                                        esrc, enorm, kEdges, total);

    // --- layer 1: relu(scatter(x @ W1) + b1) ---
    gcn_gemm_n64_wmma<<<gemmBlocks, 256, 0, stream>>>(x, W1, bufA, kNodes, kInDim);
    gcn_agg_csr<<<aggBlocks, 256, 0, stream>>>(rowptr, esrc, enorm, bufA, b1, bufB, kNodes, 1);

    // --- layer 2 ---
    gcn_gemm_n64_wmma<<<gemmBlocks, 256, 0, stream>>>(bufB, W2, bufA, kNodes, kHid);
    gcn_agg_csr<<<aggBlocks, 256, 0, stream>>>(rowptr, esrc, enorm, bufA, b2, bufB, kNodes, 1);

    // --- layer 3 (no relu) ---
    gcn_gemm_n64_wmma<<<gemmBlocks, 256, 0, stream>>>(bufB, W3, bufA, kNodes, kHid);
    gcn_agg_csr<<<aggBlocks, 256, 0, stream>>>(rowptr, esrc, enorm, bufA, b3, bufB, kNodes, 0);

    // --- global add pool + head ---
    gcn_fill_f32<<<cdiv(kGraphs * kHid, 256), 256, 0, stream>>>(pooled, 0.0f, kGraphs * kHid);
    gcn_pool_seg<<<poolBlocks, 256, 0, stream>>>(bufB, bat, pooled, kNodes);
    gcn_head    <<<1, 128, 0, stream>>>(pooled, Wl, bl, out, kGraphs);
}